// CombinedCSA_35923106463991
// MI455X (gfx1250) — compile-verified
//
#include <hip/hip_runtime.h>
#include <math.h>

typedef float v2f __attribute__((ext_vector_type(2)));
typedef float v4f __attribute__((ext_vector_type(4)));
typedef float v8f __attribute__((ext_vector_type(8)));

#define B_   16
#define C_   256
#define HW_  16384   // 128*128
#define W_   128
#define H_   128
#define NBCH 4       // batches per chunk (64 MB x-slice -> L2 resident)

// ---------------------------------------------------------------------------
// Kernel 1: per-(b,c) max + mean over 16384 pixels -> h[b*C + c] = max + mean
// One block per (b,c) within the chunk. Streams chunk's x slice (fills L2).
// ---------------------------------------------------------------------------
__global__ __launch_bounds__(256) void csa_pool(const float* __restrict__ x,
                                                float* __restrict__ hbuf,
                                                int bc0) {
    int bc = bc0 + blockIdx.x;
    const v4f* p = (const v4f*)(x + (size_t)bc * HW_);
    int t = threadIdx.x;
    float mx = -__builtin_inff();
    float sm = 0.0f;
#pragma unroll 4
    for (int i = t; i < HW_ / 4; i += 256) {
        v4f v = p[i];
        mx = fmaxf(mx, fmaxf(fmaxf(v.x, v.y), fmaxf(v.z, v.w)));
        sm += (v.x + v.y) + (v.z + v.w);
    }
    __shared__ float smx[256];
    __shared__ float ssm[256];
    smx[t] = mx; ssm[t] = sm;
    __syncthreads();
    for (int s = 128; s > 0; s >>= 1) {
        if (t < s) { smx[t] = fmaxf(smx[t], smx[t + s]); ssm[t] += ssm[t + s]; }
        __syncthreads();
    }
    if (t == 0) hbuf[bc] = smx[0] + ssm[0] * (1.0f / (float)HW_);
}

// ---------------------------------------------------------------------------
// Kernel 2: MLP on one wave via V_WMMA_F32_16X16X4_F32 (full fp32 precision).
// fc1: out[b,r] = sum_c h[b,c]*w_fc1[r,c]  -> 64 WMMAs (K=256)
// relu -> LDS (D-layout -> row-major) -> reload in A-layout
// fc2: out[b,c] = sum_r relu[b,r]*w_fc2[c,r] -> 16 tiles x 4 WMMAs
// Computes the full 16-row tile; stores sigmoid(scale) only for rows
// [b0, b0+NBCH) (other h rows may be stale/garbage but rows are independent).
// Operand mapping (ISA 7.12.2):
//   A(m,k): lane = m + 16*(k>=2), vgpr = k%2   -> contiguous float2 loads
//   B(k,n): lane = n + 16*(k>=2), vgpr = k%2   -> contiguous float2 loads
//   D(m,n): m = vgpr + 8*(lane>=16), n = lane%16
// ---------------------------------------------------------------------------
__global__ __launch_bounds__(32) void csa_mlp(const float* __restrict__ h,
                                              const float* __restrict__ w_fc1,
                                              const float* __restrict__ w_fc2,
                                              float* __restrict__ scale,
                                              int b0) {
    __shared__ float relu_lds[16 * 16];
    const int lane = threadIdx.x;      // 0..31, full wave, EXEC all ones
    const int m    = lane & 15;        // row (A) / col (B,D)
    const int half = lane >> 4;        // selects k-pair {0,1} vs {2,3}

    // ---- fc1: A = h[16x256], B[k=c][n=r] = w_fc1[r*256 + c] ----
    v8f acc = {};
#pragma unroll 4
    for (int kk = 0; kk < C_; kk += 4) {
        int k = kk + 2 * half;
        v2f a = *(const v2f*)(h     + m * C_ + k);
        v2f b = *(const v2f*)(w_fc1 + m * C_ + k);
        acc = __builtin_amdgcn_wmma_f32_16x16x4_f32(
            false, a, false, b, (short)0, acc, false, false);
    }
#pragma unroll
    for (int v = 0; v < 8; ++v) {
        int row = v + 8 * half;                      // b
        float r = acc[v] > 0.0f ? acc[v] : 0.0f;     // ReLU
        relu_lds[row * 16 + m] = r;                  // col r = m
    }
    __syncthreads();

    // ---- fc2: A = relu[16x16], B[k=r][n=c] = w_fc2[c*16 + r] ----
    for (int ct = 0; ct < 16; ++ct) {
        v8f acc2 = {};
#pragma unroll
        for (int kk = 0; kk < 16; kk += 4) {
            int k = kk + 2 * half;
            v2f a = *(const v2f*)(relu_lds + m * 16 + k);
            v2f b = *(const v2f*)(w_fc2 + (ct * 16 + m) * 16 + k);
            acc2 = __builtin_amdgcn_wmma_f32_16x16x4_f32(
                false, a, false, b, (short)0, acc2, false, false);
        }
#pragma unroll
        for (int v = 0; v < 8; ++v) {
            int b = v + 8 * half;
            if (b >= b0 && b < b0 + NBCH) {
                int c = ct * 16 + m;
                scale[b * C_ + c] = 1.0f / (1.0f + expf(-acc2[v]));
            }
        }
    }
}

// ---------------------------------------------------------------------------
// Kernel 3: spatial stats of x*scale for the chunk. Per pixel p of batch b:
//   cat[b][0][p] = max_c(x[b,c,p]*scale[b,c]), cat[b][1][p] = mean_c(...)
// 16 blocks/batch; each thread handles 4 consecutive pixels (float4).
// x slice is L2-resident from csa_pool -> near-zero HBM traffic here.
// ---------------------------------------------------------------------------
__global__ __launch_bounds__(256) void csa_spatial(const float* __restrict__ x,
                                                   const float* __restrict__ scale,
                                                   float* __restrict__ cat,
                                                   int b0) {
    int b = b0 + (blockIdx.x >> 4);               // 16 blocks per batch
    int p = ((blockIdx.x & 15) << 10) + threadIdx.x * 4;
    __shared__ float s_scale[C_];
    s_scale[threadIdx.x] = scale[b * C_ + threadIdx.x];
    __syncthreads();
    const float* xb = x + (size_t)b * C_ * HW_ + p;
    float ninf = -__builtin_inff();
    v4f mx = { ninf, ninf, ninf, ninf };
    v4f sm = {};
#pragma unroll 4
    for (int c = 0; c < C_; ++c) {
        float s = s_scale[c];
        v4f v = *(const v4f*)(xb + (size_t)c * HW_);
        v.x *= s; v.y *= s; v.z *= s; v.w *= s;
        mx.x = fmaxf(mx.x, v.x); mx.y = fmaxf(mx.y, v.y);
        mx.z = fmaxf(mx.z, v.z); mx.w = fmaxf(mx.w, v.w);
        sm.x += v.x; sm.y += v.y; sm.z += v.z; sm.w += v.w;
    }
    const float inv = 1.0f / (float)C_;
    sm.x *= inv; sm.y *= inv; sm.z *= inv; sm.w *= inv;
    *(v4f*)(cat + (size_t)b * 2 * HW_ + p)       = mx;  // max channel
    *(v4f*)(cat + (size_t)b * 2 * HW_ + HW_ + p) = sm;  // avg channel
}

// ---------------------------------------------------------------------------
// Kernel 4: 7x7 conv (pad 3) over cat[.,2,H,W] + sigmoid -> attn[.,H,W].
// cat slice is ~512 KB (L2 resident); compute trivial.
// ---------------------------------------------------------------------------
__global__ __launch_bounds__(256) void csa_conv(const float* __restrict__ cat,
                                                const float* __restrict__ wconv,
                                                float* __restrict__ attn,
                                                int idx0) {
    __shared__ float wk[98];
    if (threadIdx.x < 98) wk[threadIdx.x] = wconv[threadIdx.x];
    __syncthreads();
    int idx = idx0 + blockIdx.x * 256 + threadIdx.x;  // b*HW + h*W + w
    int b  = idx >> 14;
    int hw = idx & (HW_ - 1);
    int h  = hw >> 7;
    int w  = hw & (W_ - 1);
    const float* c0 = cat + (size_t)b * 2 * HW_;      // max channel
    const float* c1 = c0 + HW_;                       // avg channel
    float acc = 0.0f;
#pragma unroll
    for (int dy = 0; dy < 7; ++dy) {
        int yy = h + dy - 3;
        if (yy < 0 || yy >= H_) continue;
#pragma unroll
        for (int dx = 0; dx < 7; ++dx) {
            int xx = w + dx - 3;
            if (xx < 0 || xx >= W_) continue;
            int off = yy * W_ + xx;
            acc = fmaf(c0[off], wk[dy * 7 + dx],
                  fmaf(c1[off], wk[49 + dy * 7 + dx], acc));
        }
    }
    attn[idx] = 1.0f / (1.0f + expf(-acc));
}

// ---------------------------------------------------------------------------
// Kernel 5: out = x * scale[b,c] * attn[b,p] for the chunk.
// x read is the 3rd touch (L2 hit) -> non-temporal (last use).
// out stores non-temporal: never re-read, keep L2 free for next chunk.
// ---------------------------------------------------------------------------
__global__ __launch_bounds__(256) void csa_final(const float* __restrict__ x,
                                                 const float* __restrict__ scale,
                                                 const float* __restrict__ attn,
                                                 float* __restrict__ out,
                                                 int bc0) {
    int bc    = bc0 + (blockIdx.x >> 4);          // (b*C + c)
    int chunk = blockIdx.x & 15;                  // 16 chunks of 1024 floats
    int b     = bc >> 8;
    float s   = scale[bc];
    size_t base  = (size_t)bc * HW_ + chunk * 1024 + threadIdx.x * 4;
    size_t abase = (size_t)b * HW_ + chunk * 1024 + threadIdx.x * 4;
    v4f xv = __builtin_nontemporal_load((const v4f*)(x + base));
    v4f av = *(const v4f*)(attn + abase);
    v4f o;
    o.x = xv.x * s * av.x;
    o.y = xv.y * s * av.y;
    o.z = xv.z * s * av.z;
    o.w = xv.w * s * av.w;
    __builtin_nontemporal_store(o, (v4f*)(out + base));
}

// ---------------------------------------------------------------------------
// Chunked pipeline: 4 chunks x 4 batches. Each chunk's 64 MB x-slice stays
// L2-resident (192 MB) across its pool -> spatial -> final touches, cutting
// HBM traffic from ~1.03 GB to ~0.51 GB (~22 us at 23.3 TB/s).
// ---------------------------------------------------------------------------
extern "C" void kernel_launch(void* const* d_in, const int* in_sizes, int n_in,
                              void* d_out, int out_size, void* d_ws, size_t ws_size,
                              hipStream_t stream) {
    const float* x      = (const float*)d_in[0];   // [16,256,128,128]
    const float* w_fc1  = (const float*)d_in[1];   // [16,256]
    const float* w_fc2  = (const float*)d_in[2];   // [256,16]
    const float* w_conv = (const float*)d_in[3];   // [1,2,7,7]
    float* out = (float*)d_out;

    // Workspace (floats): h[4096] | scale[4096] | cat[B*2*HW] | attn[B*HW]
    float* ws    = (float*)d_ws;
    float* hbuf  = ws;                                // 4096
    float* scale = ws + 4096;                         // 4096
    float* cat   = ws + 8192;                         // 524288
    float* attn  = ws + 8192 + (size_t)B_ * 2 * HW_;  // 262144

    for (int ck = 0; ck < B_ / NBCH; ++ck) {
        int b0  = ck * NBCH;
        int bc0 = b0 * C_;
        // 1) channel pooling for chunk: NBCH*C blocks
        csa_pool<<<NBCH * C_, 256, 0, stream>>>(x, hbuf, bc0);
        // 2) WMMA MLP + sigmoid (stores rows [b0, b0+NBCH))
        csa_mlp<<<1, 32, 0, stream>>>(hbuf, w_fc1, w_fc2, scale, b0);
        // 3) spatial max/mean of x*scale: 16 blocks per batch
        csa_spatial<<<NBCH * 16, 256, 0, stream>>>(x, scale, cat, b0);
        // 4) 7x7 conv + sigmoid
        csa_conv<<<(NBCH * HW_) / 256, 256, 0, stream>>>(cat, w_conv, attn,
                                                         b0 * HW_);
        // 5) final elementwise (NT loads/stores)
        csa_final<<<NBCH * C_ * 16, 256, 0, stream>>>(x, scale, attn, out, bc0);
    }
}